// SDGraphEncoder_16664473108567
// MI455X (gfx1250) — compile-verified
//
#include <hip/hip_runtime.h>
#include <hip/hip_bf16.h>
#include <cstddef>

// ---------------------------------------------------------------------------
// CDNA5 (gfx1250) fused SDGraphEncoder forward.
// - All channel contractions on V_WMMA_F32_16X16X4_F32 (exact fp32).
// - Graph features built in LDS (paired-K layout -> b64 LDS operand loads).
// - KNN Gram columns and edge-conv W2 staged into LDS by the Tensor Data
//   Mover (2D strided tiles, double-buffered), synced with s_wait_tensorcnt.
// ---------------------------------------------------------------------------

typedef float        v2f __attribute__((ext_vector_type(2)));
typedef float        v8f __attribute__((ext_vector_type(8)));
typedef int          v4i __attribute__((ext_vector_type(4)));
typedef int          v8i __attribute__((ext_vector_type(8)));
typedef unsigned int v4u __attribute__((ext_vector_type(4)));

#if defined(__HIP_DEVICE_COMPILE__) && __has_builtin(__builtin_amdgcn_tensor_load_to_lds)
#define USE_TDM 1
#else
#define USE_TDM 0
#endif

__device__ __forceinline__ v8f wmma4(v2f a, v2f b, v8f c) {
    return __builtin_amdgcn_wmma_f32_16x16x4_f32(
        false, a, false, b, (short)0, c, false, false);
}

__device__ __forceinline__ float gelu_exact(float x) {
    return 0.5f * x * (1.0f + erff(x * 0.70710678118654752f));
}

#if USE_TDM
// TDM load of a 2D tile: d1 rows of d0 contiguous dwords, row stride
// `stride0` dwords in global, packed row-major into LDS at dstLds.
// D# per CDNA5 ISA ch.8; this toolchain's builtin takes 6 args.
__device__ __forceinline__ void tdm_load_2d(const float* src, float* dstLds,
                                            unsigned int d0, unsigned int d1,
                                            unsigned int stride0) {
    unsigned long long ga = (unsigned long long)(const void*)src;
    unsigned int lds = (unsigned int)(unsigned long long)(void*)dstLds;
    v4u g0;
    g0.x = 1u;                                            // count=1, user desc
    g0.y = lds;                                           // lds_addr (bytes)
    g0.z = (unsigned int)ga;                              // global_addr[31:0]
    g0.w = (unsigned int)((ga >> 32) & 0x1FFFFFFu) | (2u << 30);  // [56:32] | type=2
    v8i g1 = {
        (int)(2u << 16),                                  // data_size = 4B
        (int)((d0 & 0xFFFFu) << 16),                      // tensor_dim0[15:0] @63:48
        (int)(((d0 >> 16) & 0xFFFFu) | ((d1 & 0xFFFFu) << 16)),  // dim0 hi | tensor_dim1 lo
        (int)(((d1 >> 16) & 0xFFFFu) | ((d0 & 0xFFFFu) << 16)),  // dim1 hi | tile_dim0
        (int)(d1 & 0xFFFFu),                              // tile_dim1 | tile_dim2=0
        (int)stride0,                                     // tensor_dim0_stride[31:0]
        0, 0
    };
    v4i gz4 = {0, 0, 0, 0};
    v8i gz8 = {0, 0, 0, 0, 0, 0, 0, 0};
    __builtin_amdgcn_tensor_load_to_lds(g0, g1, gz4, gz4, gz8, 0);
}
#endif

// ---------------------------------------------------------------------------
// Weight prep: fold BN into weights/bias, zero-pad to WMMA tiles.
// ---------------------------------------------------------------------------
__global__ void k_prep(const float* __restrict__ W, const float* __restrict__ b,
                       const float* __restrict__ g, const float* __restrict__ be,
                       float* __restrict__ Wp, float* __restrict__ bp,
                       int O, int C, int Opad, int Cpad)
{
    int i = blockIdx.x * 256 + threadIdx.x;
    int tot = Opad * Cpad;
    if (i < tot) {
        int o = i / Cpad, c = i % Cpad;
        Wp[i] = (o < O && c < C) ? g[o] * W[o * C + c] : 0.0f;
    }
    if (i < Opad) bp[i] = (i < O) ? g[i] * b[i] + be[i] : 0.0f;
}

// ---------------------------------------------------------------------------
// d2s: 1x3 conv (stride1,pad0,32ch) + BN + GELU + max over 30 positions.
// ---------------------------------------------------------------------------
__global__ void k_d2s(const float* __restrict__ dense, const float* __restrict__ Wp,
                      const float* __restrict__ bp, float* __restrict__ spd)
{
    __shared__ float xs[32 * 32];
    int bb = blockIdx.x, s = blockIdx.y;
    for (int e = threadIdx.x; e < 1024; e += 32)
        xs[e] = dense[(size_t)bb * 32768 + (size_t)(e >> 5) * 1024 + s * 32 + (e & 31)];
    __syncthreads();
    int o = threadIdx.x;
    float mx = -3.0e38f;
    for (int t = 0; t < 30; ++t) {
        float acc = bp[o];
        for (int c = 0; c < 32; ++c) {
            const float* w = Wp + o * 96 + c * 3;
            acc += w[0] * xs[c * 32 + t] + w[1] * xs[c * 32 + t + 1] + w[2] * xs[c * 32 + t + 2];
        }
        mx = fmaxf(mx, gelu_exact(acc));
    }
    spd[((size_t)bb * 32 + o) * 32 + s] = mx;
}

// ---------------------------------------------------------------------------
// Channel-concat builders (stride-64 layouts).
// ---------------------------------------------------------------------------
__global__ void k_union_sp(const float* __restrict__ sf, const float* __restrict__ spd,
                           float* __restrict__ U)
{
    int i = blockIdx.x * 256 + threadIdx.x;
    if (i >= 32 * 64 * 32) return;
    int bb = i / (64 * 32), r = i % (64 * 32), c = r >> 5, n = r & 31;
    U[i] = (c < 32) ? sf[((size_t)bb * 32 + c) * 32 + n]
                    : spd[((size_t)bb * 32 + (c - 32)) * 32 + n];
}

__global__ void k_union_dn(const float* __restrict__ df, const float* __restrict__ sf,
                           float* __restrict__ U)
{
    int i = blockIdx.x * 256 + threadIdx.x;
    if (i >= 32 * 64 * 1024) return;
    int bb = i / (64 * 1024), r = i % (64 * 1024), c = r >> 10, n = r & 1023;
    U[i] = (c < 32) ? df[((size_t)bb * 32 + c) * 1024 + n]
                    : sf[((size_t)bb * 32 + (c - 32)) * 32 + (n >> 5)];
}

// ---------------------------------------------------------------------------
// xx[b,n] = sum_c x[b,c,n]^2   (pad rows are zero)
// ---------------------------------------------------------------------------
__global__ void k_xx(const float* __restrict__ x, int Cpad, int N, float* __restrict__ xx)
{
    int bb = blockIdx.x;
    for (int n = threadIdx.x; n < N; n += blockDim.x) {
        const float* p = x + (size_t)bb * Cpad * N + n;
        float s = 0.0f;
        for (int c = 0; c < Cpad; ++c) { float v = p[(size_t)c * N]; s += v * v; }
        xx[bb * N + n] = s;
    }
}

// ---------------------------------------------------------------------------
// KNN: per (batch, 16-row tile).  A-tile staged to LDS (paired-K); column
// panels TDM-streamed into a double-buffered LDS tile (256 cols x Cpad rows,
// row stride N); WMMA Gram fed entirely from LDS.  pdist tile in LDS ->
// iterative wave-argmax top-K (ties -> lower index).  block 256 (8 waves).
// dyn LDS = (16*N + 16*Cpad + 2*Cpad*CC)*4, CC = min(N,256).
// ---------------------------------------------------------------------------
__global__ void k_knn(const float* __restrict__ x, const float* __restrict__ xx,
                      int* __restrict__ idx, int N, int Cpad, int Kn)
{
    extern __shared__ float sm[];
    const int CC = (N < 256) ? N : 256;
    float* pd  = sm;                       // [16][N]
    float* Ap  = sm + 16 * N;              // paired-K A tile [Cpad/2][16][2]
    float* Bp0 = Ap + 16 * Cpad;           // [Cpad][CC]
    float* Bp1 = Bp0 + Cpad * CC;          // [Cpad][CC]
    const int bb = blockIdx.x, rt = blockIdx.y;
    const int row0 = rt * 16;
    const int tid = threadIdx.x;
    const int lane = tid & 31, wave = tid >> 5;
    const int m = lane & 15, kh = lane >> 4;
    const float* xb = x + (size_t)bb * Cpad * N;
    const int nchunks = N / CC;

    // Stage A tile (rows row0..row0+15, all Cpad channels) in paired-K layout.
    for (int e = tid; e < 16 * Cpad; e += 256) {
        int k = e / 16, mm = e % 16;
        Ap[(((k >> 1) * 16 + mm) << 1) | (k & 1)] = xb[(size_t)k * N + row0 + mm];
    }
    // Kick off first column panel.
#if USE_TDM
    if (wave == 0) tdm_load_2d(xb, Bp0, (unsigned)CC, (unsigned)Cpad, (unsigned)N);
#else
    for (int e = tid; e < Cpad * CC; e += 256)
        Bp0[e] = xb[(size_t)(e / CC) * N + (e % CC)];
#endif

    for (int ch = 0; ch < nchunks; ++ch) {
        float* Bcur = (ch & 1) ? Bp1 : Bp0;
        float* Bnxt = (ch & 1) ? Bp0 : Bp1;
        const int col0 = ch * CC;
#if USE_TDM
        if (wave == 0) {
            if (ch + 1 < nchunks) {
                tdm_load_2d(xb + (col0 + CC), Bnxt, (unsigned)CC, (unsigned)Cpad, (unsigned)N);
                __builtin_amdgcn_s_wait_tensorcnt(1);   // retire panel `ch`
            } else {
                __builtin_amdgcn_s_wait_tensorcnt(0);
            }
        }
#else
        if (ch + 1 < nchunks)
            for (int e = tid; e < Cpad * CC; e += 256)
                Bnxt[e] = xb[(size_t)(e / CC) * N + col0 + CC + (e % CC)];
#endif
        __syncthreads();

        for (int ct = wave; ct < CC / 16; ct += 8) {
            const int c16 = ct * 16;
            const v2f* pa = ((const v2f*)Ap) + (size_t)kh * 16 + m;
            const float* pb = Bcur + (size_t)(2 * kh) * CC + c16 + m;
            v8f acc = {};
            for (int k0 = 0; k0 < Cpad; k0 += 4) {
                v2f a = pa[0], bv;
                bv.x = pb[0]; bv.y = pb[CC];
                acc = wmma4(a, bv, acc);
                pa += 32; pb += 4 * CC;
            }
#pragma unroll
            for (int r = 0; r < 8; ++r) {
                int ml = r + 8 * kh;
                pd[ml * N + col0 + c16 + m] =
                    2.0f * acc[r] - xx[bb * N + row0 + ml] - xx[bb * N + col0 + c16 + m];
            }
        }
        __syncthreads();
    }

    // Top-K: wave handles rows 2*wave, 2*wave+1.
    const int cnt = N >> 5;
    float vals[32];
    for (int rr = 0; rr < 2; ++rr) {
        const int rl = wave * 2 + rr;
        for (int j = 0; j < cnt; ++j) vals[j] = pd[rl * N + j * 32 + lane];
        for (int it = 0; it < Kn; ++it) {
            float bv = -3.0e38f; int bj = 0;
            for (int j = 0; j < cnt; ++j)
                if (vals[j] > bv) { bv = vals[j]; bj = j; }
            int bi = bj * 32 + lane;
            for (int off = 16; off > 0; off >>= 1) {
                float ov = __shfl_xor(bv, off, 32);
                int   oi = __shfl_xor(bi, off, 32);
                if (ov > bv || (ov == bv && oi < bi)) { bv = ov; bi = oi; }
            }
            if ((bi & 31) == lane) vals[bi >> 5] = -3.0e38f;
            if (lane == 0) idx[((size_t)bb * N + row0 + rl) * Kn + it] = bi;
        }
    }
}

// ---------------------------------------------------------------------------
// Fused edge-conv: gather [nbr-ctr, ctr] -> WMMA GEMM1 -> GELU -> WMMA GEMM2
// -> GELU -> max over K=10.  8 points/block (80 cols = 5 tiles), block 256.
// LDS (paired-K): F[CFPAD/2][80][2], Y1[O1PAD/2][80][2]; W2 TDM-staged.
// ---------------------------------------------------------------------------
template <int CFPAD, int O1PAD, int O2PAD>
__global__ __launch_bounds__(256)
void k_graph_mlp(const float* __restrict__ x, int Cstride, int C,
                 const int* __restrict__ idx,
                 const float* __restrict__ W1, const float* __restrict__ b1,
                 const float* __restrict__ W2, const float* __restrict__ b2,
                 float* __restrict__ out, int N)
{
    constexpr int COLS = 80, NT = 8, KNN = 10, CT = COLS / 16;
    extern __shared__ float smem[];
    float* F   = smem;                                  // CFPAD*COLS (paired-K)
    float* Y1  = smem + CFPAD * COLS;                   // O1PAD*COLS (paired-K)
    float* W2l = smem + (CFPAD + O1PAD) * COLS;         // O2PAD*O1PAD row-major
    const int bb = blockIdx.x, pt = blockIdx.y;
    const int n0 = pt * NT;
    const float* xb = x + (size_t)bb * Cstride * N;
    const int tid = threadIdx.x;
    const int lane = tid & 31, wave = tid >> 5;
    const int m = lane & 15, kh = lane >> 4;

#if USE_TDM
    if (wave == 0)
        tdm_load_2d(W2, W2l, (unsigned)(O2PAD * O1PAD), 1u, (unsigned)(O2PAD * O1PAD));
#endif

    // Phase A: gather graph features into paired-K LDS (+ prefetch W1)
    for (int e = tid * 32; e < O1PAD * CFPAD; e += 256 * 32)
        __builtin_prefetch(W1 + e, 0, 3);
    for (int e = tid; e < CFPAD * COLS; e += 256) {
        int c = e / COLS, j = e % COLS;
        int p = j / KNN, kk = j % KNN;
        int n = n0 + p;
        float v = 0.0f;
        if (c < C) {
            int nb = idx[((size_t)bb * N + n) * KNN + kk];
            v = xb[(size_t)c * N + nb] - xb[(size_t)c * N + n];
        } else if (c < 2 * C) {
            v = xb[(size_t)(c - C) * N + n];
        }
        F[(((c >> 1) * COLS + j) << 1) | (c & 1)] = v;
    }
#if !USE_TDM
    for (int e = tid; e < O2PAD * O1PAD; e += 256) W2l[e] = W2[e];
#endif
    __syncthreads();

    // Phase B: GEMM1 (O1PAD x CFPAD) @ F -> GELU -> Y1 (paired-K)
    for (int t = wave; t < (O1PAD / 16) * CT; t += 8) {
        int tm = t / CT, tn = t % CT;
        const v2f* pw = (const v2f*)(W1 + (size_t)(tm * 16 + m) * CFPAD) + kh;
        const v2f* pf = ((const v2f*)F) + (size_t)kh * COLS + tn * 16 + m;
        v8f acc = {};
        for (int j = 0; j < CFPAD / 4; ++j) {
            acc = wmma4(pw[0], pf[0], acc);
            pw += 2; pf += 2 * COLS;
        }
#pragma unroll
        for (int r = 0; r < 8; ++r) {
            int ml = tm * 16 + r + 8 * kh;
            Y1[(((ml >> 1) * COLS + tn * 16 + m) << 1) | (ml & 1)] =
                gelu_exact(acc[r] + b1[ml]);
        }
    }
#if USE_TDM
    if (wave == 0) __builtin_amdgcn_s_wait_tensorcnt(0);
#endif
    __syncthreads();

    // Phase C: GEMM2 (O2PAD x O1PAD) @ Y1 -> GELU -> Y2 (reuses F, paired-K)
    float* Y2 = F;
    for (int t = wave; t < (O2PAD / 16) * CT; t += 8) {
        int tm = t / CT, tn = t % CT;
        const v2f* pw = (const v2f*)(W2l + (size_t)(tm * 16 + m) * O1PAD) + kh;
        const v2f* pf = ((const v2f*)Y1) + (size_t)kh * COLS + tn * 16 + m;
        v8f acc = {};
        for (int j = 0; j < O1PAD / 4; ++j) {
            acc = wmma4(pw[0], pf[0], acc);
            pw += 2; pf += 2 * COLS;
        }
#pragma unroll
        for (int r = 0; r < 8; ++r) {
            int ml = tm * 16 + r + 8 * kh;
            Y2[(((ml >> 1) * COLS + tn * 16 + m) << 1) | (ml & 1)] =
                gelu_exact(acc[r] + b2[ml]);
        }
    }
    __syncthreads();

    // Phase D: max over the 10 neighbor columns per point
    for (int e = tid; e < O2PAD * NT; e += 256) {
        int o = e / NT, p = e % NT;
        float mx = -3.0e38f;
        for (int kk = 0; kk < KNN; ++kk)
            mx = fmaxf(mx, Y2[(((o >> 1) * COLS + p * KNN + kk) << 1) | (o & 1)]);
        out[((size_t)bb * O2PAD + o) * N + n0 + p] = mx;
    }
}

// ---------------------------------------------------------------------------
// c3: fused 2-layer 1D MLP on concat(x1[90], x2[128]) -> 167 -> 128.
// 16 points per block, paired-K LDS, block 256.
// ---------------------------------------------------------------------------
__global__ __launch_bounds__(256)
void k_c3(const float* __restrict__ x1, const float* __restrict__ x2,
          const float* __restrict__ W1, const float* __restrict__ b1,
          const float* __restrict__ W2, const float* __restrict__ b2,
          float* __restrict__ out, int N)
{
    __shared__ float F[224 * 16];                       // paired-K
    __shared__ float Y[176 * 16];                       // paired-K
    const int bb = blockIdx.x, nt = blockIdx.y;
    const int n0 = nt * 16;
    const int tid = threadIdx.x;
    const int lane = tid & 31, wave = tid >> 5;
    const int m = lane & 15, kh = lane >> 4;

    for (int e = tid; e < 224 * 16; e += 256) {
        int c = e / 16, j = e % 16, n = n0 + j;
        float v = 0.0f;
        if (c < 90)       v = x1[((size_t)bb * 96 + c) * N + n];
        else if (c < 218) v = x2[((size_t)bb * 128 + (c - 90)) * N + n];
        F[(((c >> 1) * 16 + j) << 1) | (c & 1)] = v;
    }
    __syncthreads();

    for (int t = wave; t < 11; t += 8) {
        const v2f* pw = (const v2f*)(W1 + (size_t)(t * 16 + m) * 224) + kh;
        const v2f* pf = ((const v2f*)F) + (size_t)kh * 16 + m;
        v8f acc = {};
        for (int j = 0; j < 56; ++j) {
            acc = wmma4(pw[0], pf[0], acc);
            pw += 2; pf += 32;
        }
#pragma unroll
        for (int r = 0; r < 8; ++r) {
            int ml = t * 16 + r + 8 * kh;
            Y[(((ml >> 1) * 16 + m) << 1) | (ml & 1)] = gelu_exact(acc[r] + b1[ml]);
        }
    }
    __syncthreads();

    for (int t = wave; t < 8; t += 8) {
        const v2f* pw = (const v2f*)(W2 + (size_t)(t * 16 + m) * 176) + kh;
        const v2f* pf = ((const v2f*)Y) + (size_t)kh * 16 + m;
        v8f acc = {};
        for (int j = 0; j < 44; ++j) {
            acc = wmma4(pw[0], pf[0], acc);
            pw += 2; pf += 32;
        }
#pragma unroll
        for (int r = 0; r < 8; ++r) {
            int ml = t * 16 + r + 8 * kh;
            out[((size_t)bb * 128 + ml) * N + n0 + m] = gelu_exact(acc[r] + b2[ml]);
        }
    }
}

// ---------------------------------------------------------------------------
// down: 1x3 conv stride2 pad1 (128->128) + BN + GELU.
// ---------------------------------------------------------------------------
__global__ void k_down(const float* __restrict__ g, const float* __restrict__ Wp,
                       const float* __restrict__ bp, float* __restrict__ out)
{
    __shared__ float xs[128 * 32];
    int bb = blockIdx.x, s = blockIdx.y;
    for (int e = threadIdx.x; e < 128 * 32; e += 128)
        xs[e] = g[((size_t)bb * 128 + (e >> 5)) * 1024 + s * 32 + (e & 31)];
    __syncthreads();
    for (int e = threadIdx.x; e < 128 * 16; e += 128) {
        int o = e >> 4, t = e & 15;
        float acc = bp[o];
        int base = 2 * t - 1;
        for (int c = 0; c < 128; ++c) {
            const float* w = Wp + o * 384 + c * 3;
#pragma unroll
            for (int dt = 0; dt < 3; ++dt) {
                int tt = base + dt;
                if (tt >= 0 && tt < 32) acc += w[dt] * xs[c * 32 + tt];
            }
        }
        out[(((size_t)bb * 128 + o) * 32 + s) * 16 + t] = gelu_exact(acc);
    }
}

// ---------------------------------------------------------------------------
// Host driver
// ---------------------------------------------------------------------------
extern "C" void kernel_launch(void* const* d_in, const int* in_sizes, int n_in,
                              void* d_out, int out_size, void* d_ws, size_t ws_size,
                              hipStream_t stream)
{
    (void)in_sizes; (void)n_in; (void)out_size; (void)ws_size;
    const float* sparse_fea = (const float*)d_in[0];
    const float* dense_fea  = (const float*)d_in[1];
    float* outp = (float*)d_out;

    float* ws = (float*)d_ws;
    size_t off = 0;
    auto alloc = [&](size_t n) { float* p = ws + off; off += (n + 63) & ~(size_t)63; return p; };

    float* Wd2s = alloc(32 * 96);   float* Bd2s = alloc(32);
    struct GW { float *W1a, *B1a, *W1b, *B1b, *W2a, *B2a, *W2b, *B2b, *W3a, *B3a, *W3b, *B3b; };
    GW gw[2];
    for (int gi = 0; gi < 2; ++gi) {
        gw[gi].W1a = alloc(112 * 128); gw[gi].B1a = alloc(112);
        gw[gi].W1b = alloc(96 * 112);  gw[gi].B1b = alloc(96);
        gw[gi].W2a = alloc(160 * 192); gw[gi].B2a = alloc(160);
        gw[gi].W2b = alloc(128 * 160); gw[gi].B2b = alloc(128);
        gw[gi].W3a = alloc(176 * 224); gw[gi].B3a = alloc(176);
        gw[gi].W3b = alloc(128 * 176); gw[gi].B3b = alloc(128);
    }
    float* Wdwn = alloc(128 * 384); float* Bdwn = alloc(128);

    float* SPD  = alloc(32 * 32 * 32);
    float* Usp  = alloc(32 * 64 * 32);
    float* Udn  = alloc((size_t)32 * 64 * 1024);
    float* X1sp = alloc(32 * 96 * 32);
    float* X1dn = alloc((size_t)32 * 96 * 1024);
    float* X2sp = alloc(32 * 128 * 32);
    float* X2dn = alloc((size_t)32 * 128 * 1024);
    float* Gdn  = alloc((size_t)32 * 128 * 1024);
    float* XX   = alloc(32 * 1024);
    int*   IDX  = (int*)alloc((size_t)32 * 1024 * 10);

    auto prep = [&](int base, float* Wp, float* bp, int O, int C, int Op, int Cp) {
        int tot = Op * Cp;
        k_prep<<<dim3((tot + 255) / 256), dim3(256), 0, stream>>>(
            (const float*)d_in[base], (const float*)d_in[base + 1],
            (const float*)d_in[base + 2], (const float*)d_in[base + 3],
            Wp, bp, O, C, Op, Cp);
    };

    prep(2, Wd2s, Bd2s, 32, 96, 32, 96);
    for (int gi = 0; gi < 2; ++gi) {
        int b0 = 6 + gi * 24;
        prep(b0 + 0,  gw[gi].W1a, gw[gi].B1a, 107, 128, 112, 128);
        prep(b0 + 4,  gw[gi].W1b, gw[gi].B1b,  90, 107,  96, 112);
        prep(b0 + 8,  gw[gi].W2a, gw[gi].B2a, 151, 180, 160, 192);
        prep(b0 + 12, gw[gi].W2b, gw[gi].B2b, 128, 151, 128, 160);
        prep(b0 + 16, gw[gi].W3a, gw[gi].B3a, 167, 218, 176, 224);
        prep(b0 + 20, gw[gi].W3b, gw[gi].B3b, 128, 167, 128, 176);
    }
    prep(54, Wdwn, Bdwn, 128, 384, 128, 384);

    constexpr int SH1 = ((128 + 112) * 80 + 96 * 112) * 4;   // 119,808 B
    constexpr int SH2 = ((192 + 160) * 80 + 128 * 160) * 4;  // 194,560 B
    // KNN dyn LDS: (16*N + 16*Cpad + 2*Cpad*CC)*4
    auto knn_sh = [](int N, int Cpad) {
        int CC = (N < 256) ? N : 256;
        return (size_t)(16 * N + 16 * Cpad + 2 * Cpad * CC) * 4;
    };
    {
        auto* f1 = k_graph_mlp<128, 112, 96>;
        auto* f2 = k_graph_mlp<192, 160, 128>;
        (void)hipFuncSetAttribute((const void*)f1,
            hipFuncAttributeMaxDynamicSharedMemorySize, SH1);
        (void)hipFuncSetAttribute((const void*)f2,
            hipFuncAttributeMaxDynamicSharedMemorySize, SH2);
        (void)hipFuncSetAttribute((const void*)k_knn,
            hipFuncAttributeMaxDynamicSharedMemorySize, (int)knn_sh(1024, 96));
    }

    k_d2s<<<dim3(32, 32), dim3(32), 0, stream>>>(dense_fea, Wd2s, Bd2s, SPD);
    k_union_sp<<<dim3((32 * 64 * 32 + 255) / 256), dim3(256), 0, stream>>>(sparse_fea, SPD, Usp);
    k_union_dn<<<dim3((32 * 64 * 1024 + 255) / 256), dim3(256), 0, stream>>>(dense_fea, sparse_fea, Udn);

    auto run_gcn = [&](const float* X0, int N, const GW& w,
                       float* X1, float* X2, float* OUT) {
        k_xx<<<dim3(32), dim3(256), 0, stream>>>(X0, 64, N, XX);
        k_knn<<<dim3(32, N / 16), dim3(256), knn_sh(N, 64), stream>>>(X0, XX, IDX, N, 64, 10);
        k_graph_mlp<128, 112, 96><<<dim3(32, N / 8), dim3(256), SH1, stream>>>(
            X0, 64, 64, IDX, w.W1a, w.B1a, w.W1b, w.B1b, X1, N);
        k_xx<<<dim3(32), dim3(256), 0, stream>>>(X1, 96, N, XX);
        k_knn<<<dim3(32, N / 16), dim3(256), knn_sh(N, 96), stream>>>(X1, XX, IDX, N, 96, 10);
        k_graph_mlp<192, 160, 128><<<dim3(32, N / 8), dim3(256), SH2, stream>>>(
            X1, 96, 90, IDX, w.W2a, w.B2a, w.W2b, w.B2b, X2, N);
        k_c3<<<dim3(32, N / 16), dim3(256), 0, stream>>>(
            X1, X2, w.W3a, w.B3a, w.W3b, w.B3b, OUT, N);
    };
    run_gcn(Usp, 32,   gw[0], X1sp, X2sp, outp);
    run_gcn(Udn, 1024, gw[1], X1dn, X2dn, Gdn);

    k_down<<<dim3(32, 32), dim3(128), 0, stream>>>(Gdn, Wdwn, Bdwn, outp + 32 * 128 * 32);
}